// skew_IoULoss_67637144978187
// MI455X (gfx1250) — compile-verified
//
#include <hip/hip_runtime.h>
#include <stdint.h>

#define EPSF 1e-6f

typedef __attribute__((ext_vector_type(2))) float v2f;
typedef __attribute__((ext_vector_type(8))) float v8f;

// ---------------------------------------------------------------------------
// CDNA5 async Global -> LDS copy (ASYNCcnt path).
// ldsByteOff = offset within the workgroup's LDS allocation (low 32 bits of
// the flat shared-pointer address per ISA section 10.2 aperture mapping).
// ---------------------------------------------------------------------------
__device__ __forceinline__ void async_load_lds_b128(uint32_t ldsByteOff, const void* gptr)
{
    asm volatile("global_load_async_to_lds_b128 %0, %1, off"
                 :: "v"(ldsByteOff), "v"(gptr)
                 : "memory");
}
__device__ __forceinline__ void wait_asynccnt0()
{
    asm volatile("s_wait_asynccnt 0x0" ::: "memory");
}

// ---------------------------------------------------------------------------
// Per-pair rotated-polygon IoU loss term (mirrors reference pairwise_skew_iou)
// pp/qq: 8 floats each = x0,y0,x1,y1,x2,y2,x3,y3 ; a1/a2: shoelace areas
// ---------------------------------------------------------------------------
__device__ __forceinline__ float skew_term(const float* __restrict__ pp, float a1,
                                           const float* __restrict__ qq, float a2)
{
    float px[4], py[4], qx[4], qy[4];
#pragma unroll
    for (int i = 0; i < 4; ++i) {
        px[i] = pp[2*i]; py[i] = pp[2*i+1];
        qx[i] = qq[2*i]; qy[i] = qq[2*i+1];
    }

    float ptsx[24], ptsy[24];
    bool  val[24];

    // 16 edge-edge intersection candidates (k = i*4 + j, matching reference reshape)
#pragma unroll
    for (int i = 0; i < 4; ++i) {
        const int i1 = (i + 1) & 3;
        float rx = px[i1] - px[i], ry = py[i1] - py[i];
#pragma unroll
        for (int j = 0; j < 4; ++j) {
            const int j1 = (j + 1) & 3;
            float sx = qx[j1] - qx[j], sy = qy[j1] - qy[j];
            float den = rx * sy - ry * sx;
            bool ok = fabsf(den) > EPSF;
            float dens = ok ? den : 1.0f;
            float dqx = qx[j] - px[i], dqy = qy[j] - py[i];
            float t = (dqx * sy - dqy * sx) / dens;
            float u = (dqx * ry - dqy * rx) / dens;
            int k = i * 4 + j;
            val[k]  = ok && (t >= 0.0f) && (t <= 1.0f) && (u >= 0.0f) && (u <= 1.0f);
            ptsx[k] = px[i] + t * rx;
            ptsy[k] = py[i] + t * ry;
        }
    }

    // P corners inside Q
#pragma unroll
    for (int i = 0; i < 4; ++i) {
        bool ge = true, le = true;
#pragma unroll
        for (int j = 0; j < 4; ++j) {
            const int j1 = (j + 1) & 3;
            float c = (qx[j1] - qx[j]) * (py[i] - qy[j]) - (qy[j1] - qy[j]) * (px[i] - qx[j]);
            ge = ge && (c >= -EPSF);
            le = le && (c <=  EPSF);
        }
        ptsx[16 + i] = px[i]; ptsy[16 + i] = py[i]; val[16 + i] = ge || le;
    }
    // Q corners inside P
#pragma unroll
    for (int i = 0; i < 4; ++i) {
        bool ge = true, le = true;
#pragma unroll
        for (int j = 0; j < 4; ++j) {
            const int j1 = (j + 1) & 3;
            float c = (px[j1] - px[j]) * (qy[i] - py[j]) - (py[j1] - py[j]) * (qx[i] - px[j]);
            ge = ge && (c >= -EPSF);
            le = le && (c <=  EPSF);
        }
        ptsx[20 + i] = qx[i]; ptsy[20 + i] = qy[i]; val[20 + i] = ge || le;
    }

    // centroid of valid points
    int cnt = 0; float sx = 0.0f, sy = 0.0f;
#pragma unroll
    for (int k = 0; k < 24; ++k)
        if (val[k]) { cnt++; sx += ptsx[k]; sy += ptsy[k]; }
    float inv = 1.0f / (float)(cnt > 1 ? cnt : 1);
    float cx = sx * inv, cy = sy * inv;

    // polar angles (invalid -> 1e9, matching reference)
    float ang[24];
#pragma unroll
    for (int k = 0; k < 24; ++k)
        ang[k] = val[k] ? atan2f(ptsy[k] - cy, ptsx[k] - cx) : 1.0e9f;

    // stable bubble sort network on (ang, x, y, val) — compile-time indices only,
    // stays in VGPRs (strict > compare keeps it stable, matching jnp.argsort)
#pragma unroll
    for (int i = 0; i < 23; ++i) {
#pragma unroll
        for (int j = 0; j < 23 - i; ++j) {
            bool sw = ang[j] > ang[j + 1];
            float t;
            t = ang[j];  ang[j]  = sw ? ang[j+1]  : t; ang[j+1]  = sw ? t : ang[j+1];
            t = ptsx[j]; ptsx[j] = sw ? ptsx[j+1] : t; ptsx[j+1] = sw ? t : ptsx[j+1];
            t = ptsy[j]; ptsy[j] = sw ? ptsy[j+1] : t; ptsy[j+1] = sw ? t : ptsy[j+1];
            bool b = val[j]; val[j] = sw ? val[j+1] : b; val[j+1] = sw ? b : val[j+1];
        }
    }

    // replace invalid (sorted-last) points with first sorted point
    float fx = ptsx[0], fy = ptsy[0];
#pragma unroll
    for (int k = 0; k < 24; ++k)
        if (!val[k]) { ptsx[k] = fx; ptsy[k] = fy; }

    // shoelace of 24-gon
    float s2 = 0.0f;
#pragma unroll
    for (int k = 0; k < 24; ++k) {
        int k1 = (k + 1) % 24;
        s2 += ptsx[k] * ptsy[k1] - ptsy[k] * ptsx[k1];
    }
    float inter = 0.5f * fabsf(s2);
    float uni   = fmaxf(a1 + a2 - inter, EPSF);
    float iou   = inter / uni;
    return -logf(fmaxf(iou, EPSF));
}

// ---------------------------------------------------------------------------
// Prologue: rotbox -> corners, shoelace area, AABB
// ---------------------------------------------------------------------------
__global__ void build_polys_kernel(const float* __restrict__ boxes, int n,
                                   float* __restrict__ poly, float* __restrict__ area,
                                   float* __restrict__ aabb)
{
    int i = blockIdx.x * blockDim.x + threadIdx.x;
    if (i >= n) return;
    float xc = boxes[i*5+0], yc = boxes[i*5+1];
    float w  = boxes[i*5+2], h  = boxes[i*5+3];
    float a  = boxes[i*5+4];
    float c = cosf(a), s = sinf(a);
    const float dx[4] = {-0.5f*w,  0.5f*w, 0.5f*w, -0.5f*w};
    const float dy[4] = {-0.5f*h, -0.5f*h, 0.5f*h,  0.5f*h};
    float X[4], Y[4];
    float mnx = 1e30f, mny = 1e30f, mxx = -1e30f, mxy = -1e30f;
#pragma unroll
    for (int j = 0; j < 4; ++j) {
        X[j] = xc + dx[j]*c - dy[j]*s;
        Y[j] = yc + dx[j]*s + dy[j]*c;
        poly[i*8 + 2*j]     = X[j];
        poly[i*8 + 2*j + 1] = Y[j];
        mnx = fminf(mnx, X[j]); mxx = fmaxf(mxx, X[j]);
        mny = fminf(mny, Y[j]); mxy = fmaxf(mxy, Y[j]);
    }
    float s2 = 0.0f;
#pragma unroll
    for (int j = 0; j < 4; ++j) { int j1 = (j+1)&3; s2 += X[j]*Y[j1] - Y[j]*X[j1]; }
    area[i] = 0.5f * fabsf(s2);
    aabb[i*4+0] = mnx; aabb[i*4+1] = mny; aabb[i*4+2] = mxx; aabb[i*4+3] = mxy;
}

// ---------------------------------------------------------------------------
// Phase 1: AABB-overlap mask; zero the terms array; compact surviving pairs.
// (terms[] is index-addressed -> result independent of append order.)
// ---------------------------------------------------------------------------
__global__ void mask_compact_kernel(const float* __restrict__ aabbP, const float* __restrict__ aabbG,
                                    int N, int M, float* __restrict__ terms,
                                    int* __restrict__ list, int* __restrict__ count)
{
    int p = blockIdx.x * blockDim.x + threadIdx.x;
    if (p >= N * M) return;
    terms[p] = 0.0f;
    int n = p / M, m = p - n * M;
    float wx = fminf(aabbP[n*4+2], aabbG[m*4+2]) - fmaxf(aabbP[n*4+0], aabbG[m*4+0]);
    float wy = fminf(aabbP[n*4+3], aabbG[m*4+3]) - fmaxf(aabbP[n*4+1], aabbG[m*4+1]);
    if (wx > 0.0f && wy > 0.0f) {
        int idx = atomicAdd(count, 1);
        list[idx] = p;
    }
}

// ---------------------------------------------------------------------------
// Phase 2: heavy geometry only on surviving pairs. Target polys + areas live
// contiguously in ws (9*M floats); staged into LDS via the CDNA5 async
// Global->LDS engine when alignment permits.
// ---------------------------------------------------------------------------
__global__ void __launch_bounds__(256) skew_pairs_kernel(
        const float* __restrict__ polyP, const float* __restrict__ areaP,
        const float* __restrict__ polyG, const float* __restrict__ areaG,
        const int* __restrict__ list, const int* __restrict__ count,
        float* __restrict__ terms, int M)
{
    __shared__ float sG[256 * 9];           // [0,8M): polys, [8M,9M): areas
    const bool useLds = (M <= 256);
    if (useLds) {
        const bool asyncOk = ((M & 3) == 0) && ((((uintptr_t)polyG) & 15) == 0);
        if (asyncOk) {
            // polyG (8M floats) and areaG (M floats) are contiguous in ws.
            const int n128 = (M * 9) / 4;                     // float4 chunks
            const uint32_t ldsBase = (uint32_t)(uintptr_t)(void*)sG;
            for (int i = threadIdx.x; i < n128; i += blockDim.x)
                async_load_lds_b128(ldsBase + (uint32_t)i * 16u,
                                    (const char*)polyG + (size_t)i * 16u);
            wait_asynccnt0();               // drain this wave's ASYNCcnt
        } else {
            for (int i = threadIdx.x; i < M * 8; i += blockDim.x) sG[i] = polyG[i];
            for (int i = threadIdx.x; i < M;     i += blockDim.x) sG[M*8 + i] = areaG[i];
        }
    }
    __syncthreads();                        // all waves' staging visible
    int total  = count[0];
    int stride = gridDim.x * blockDim.x;
    for (int idx = blockIdx.x * blockDim.x + threadIdx.x; idx < total; idx += stride) {
        __builtin_prefetch(&list[idx + stride], 0, 1);   // -> global_prefetch_b8
        int p = list[idx];
        int n = p / M, m = p - (p / M) * M;
        const float* qq = useLds ? (sG + m * 8) : (polyG + m * 8);
        float a2        = useLds ? sG[M*8 + m]  : areaG[m];
        terms[p] = skew_term(polyP + n * 8, areaP[n], qq, a2);
    }
}

// ---------------------------------------------------------------------------
// Fallback (small workspace): one thread per pair with early-out, fixed-tree
// block reduction straight to partials.
// ---------------------------------------------------------------------------
__global__ void __launch_bounds__(256) skew_direct_kernel(
        const float* __restrict__ polyP, const float* __restrict__ areaP, const float* __restrict__ aabbP,
        const float* __restrict__ polyG, const float* __restrict__ areaG, const float* __restrict__ aabbG,
        float* __restrict__ partials, int N, int M)
{
    __shared__ float red[256];
    int p = blockIdx.x * 256 + threadIdx.x;
    float term = 0.0f;
    if (p < N * M) {
        int n = p / M, m = p - (p / M) * M;
        float wx = fminf(aabbP[n*4+2], aabbG[m*4+2]) - fmaxf(aabbP[n*4+0], aabbG[m*4+0]);
        float wy = fminf(aabbP[n*4+3], aabbG[m*4+3]) - fmaxf(aabbP[n*4+1], aabbG[m*4+1]);
        if (wx > 0.0f && wy > 0.0f)
            term = skew_term(polyP + n * 8, areaP[n], polyG + m * 8, areaG[m]);
    }
    red[threadIdx.x] = term;
    __syncthreads();
    for (int s = 128; s > 0; s >>= 1) {
        if (threadIdx.x < s) red[threadIdx.x] += red[threadIdx.x + s];
        __syncthreads();
    }
    if (threadIdx.x == 0) partials[blockIdx.x] = red[0];
}

// ---------------------------------------------------------------------------
// terms[] -> per-block partials (deterministic fixed tree)
// ---------------------------------------------------------------------------
__global__ void reduce_terms_kernel(const float* __restrict__ terms, int total,
                                    float* __restrict__ partials)
{
    __shared__ float red[256];
    int p = blockIdx.x * 256 + threadIdx.x;
    red[threadIdx.x] = (p < total) ? terms[p] : 0.0f;
    __syncthreads();
    for (int s = 128; s > 0; s >>= 1) {
        if (threadIdx.x < s) red[threadIdx.x] += red[threadIdx.x + s];
        __syncthreads();
    }
    if (threadIdx.x == 0) partials[blockIdx.x] = red[0];
}

// ---------------------------------------------------------------------------
// Final: single wave32. Per-lane strided sums, then V_WMMA_F32_16X16X4_F32
// with A=ones collapses all 64 B elements (2/lane) into every D column;
// 4 shfl_xor finish the 16 column sums. EXEC is all-1s (32 threads, no
// divergence at the WMMA).
// ---------------------------------------------------------------------------
__global__ void final_wmma_reduce_kernel(const float* __restrict__ partials, int nb,
                                         float* __restrict__ out)
{
    int lane = threadIdx.x;              // launched <<<1, 32>>>
    float v = 0.0f;
    for (int i = lane; i < nb; i += 32) v += partials[i];

    v2f A; A[0] = 1.0f; A[1] = 1.0f;     // 16x4 ones
    v2f B; B[0] = v;    B[1] = 0.0f;     // 4x16 holds the 32 lane sums
    v8f C = {};
    C = __builtin_amdgcn_wmma_f32_16x16x4_f32(
            /*neg_a=*/false, A, /*neg_b=*/false, B,
            /*c_mod=*/(short)0, C, /*reuse_a=*/false, /*reuse_b=*/false);

    float r = C[0];                       // D row 0: column sums across lanes 0..15
    r += __shfl_xor(r, 1, 32);
    r += __shfl_xor(r, 2, 32);
    r += __shfl_xor(r, 4, 32);
    r += __shfl_xor(r, 8, 32);
    if (lane == 0) out[0] = r;
}

// ---------------------------------------------------------------------------
extern "C" void kernel_launch(void* const* d_in, const int* in_sizes, int n_in,
                              void* d_out, int out_size, void* d_ws, size_t ws_size,
                              hipStream_t stream)
{
    (void)n_in; (void)out_size;
    const float* pred = (const float*)d_in[0];
    const float* targ = (const float*)d_in[1];
    const int N = in_sizes[0] / 5;
    const int M = in_sizes[1] / 5;
    const long long total = (long long)N * (long long)M;
    const int NBLK = (int)((total + 255) / 256);

    float* ws = (float*)d_ws;
    size_t o = 0;
    float* polyP    = ws + o; o += (size_t)8 * N;
    float* areaP    = ws + o; o += (size_t)N;
    float* aabbP    = ws + o; o += (size_t)4 * N;
    float* polyG    = ws + o; o += (size_t)8 * M;   // NOTE: areaG must follow
    float* areaG    = ws + o; o += (size_t)M;       //       polyG contiguously
    float* aabbG    = ws + o; o += (size_t)4 * M;
    float* partials = ws + o; o += (size_t)NBLK;
    int*   count    = (int*)(ws + o); o += 1;
    int*   list     = (int*)(ws + o); o += (size_t)total;
    float* terms    = ws + o;         o += (size_t)total;
    const size_t compact_bytes = o * sizeof(float);

    build_polys_kernel<<<(N + 255) / 256, 256, 0, stream>>>(pred, N, polyP, areaP, aabbP);
    build_polys_kernel<<<(M + 255) / 256, 256, 0, stream>>>(targ, M, polyG, areaG, aabbG);

    if (ws_size >= compact_bytes) {
        hipMemsetAsync(count, 0, sizeof(int), stream);
        mask_compact_kernel<<<NBLK, 256, 0, stream>>>(aabbP, aabbG, N, M, terms, list, count);
        int blocks2 = NBLK < 512 ? NBLK : 512;
        skew_pairs_kernel<<<blocks2, 256, 0, stream>>>(polyP, areaP, polyG, areaG,
                                                       list, count, terms, M);
        reduce_terms_kernel<<<NBLK, 256, 0, stream>>>(terms, (int)total, partials);
    } else {
        skew_direct_kernel<<<NBLK, 256, 0, stream>>>(polyP, areaP, aabbP,
                                                     polyG, areaG, aabbG, partials, N, M);
    }
    final_wmma_reduce_kernel<<<1, 32, 0, stream>>>(partials, NBLK, (float*)d_out);
}